// bicubic_imresize_23407571763508
// MI455X (gfx1250) — compile-verified
//
#include <hip/hip_runtime.h>

typedef __attribute__((ext_vector_type(2))) float v2f;
typedef __attribute__((ext_vector_type(8))) float v8f;
typedef __attribute__((ext_vector_type(4))) int   v4i;
typedef v4i __attribute__((address_space(1)))* gv4i_p;   // global-AS vector ptr
typedef v4i __attribute__((address_space(3)))* lv4i_p;   // LDS-AS vector ptr

#define IN_H   1024
#define IN_W   1024
#define OUT_H  256
#define OUT_W  256
#define NIMG   24          // 8 batch * 3 channels
#define TILE   16
#define PROWS  80          // 20 K-chunks of 4 (row window 76 + 4 clamped pad rows)
#define PCOLS  84          // aligned window: c0a = 4*tw0-8, cols c0a..c0a+83 (shift +3)
#define NVEC   (PCOLS/4)   // 21 float4 per patch row
#define TPAD   84          // tmp row stride (84 mod 64 = 20 -> conflict-free column reads)
#define KOFF   60          // kpad[j - 4m + KOFF] == K[j-4m] when in [0,16)
#define KPAD_N 144

#if defined(__gfx1250__) && __has_builtin(__builtin_amdgcn_global_load_async_to_lds_b128)
#define USE_ASYNC_LDS 1
#else
#define USE_ASYNC_LDS 0
#endif

__device__ __forceinline__ int clampi(int x, int lo, int hi) {
    return x < lo ? lo : (x > hi ? hi : x);
}

__global__ __launch_bounds__(256)
void bicubic_ds4_wmma(const float* __restrict__ in, float* __restrict__ out)
{
    __shared__ float patch[PROWS][PCOLS];   // 26.9 KB input tile (all real clamped data)
    __shared__ float tmp[TILE][TPAD];       // 5.4 KB  row-pass result (padded stride)
    __shared__ float kpad[KPAD_N];          // zero-padded 16-tap filter band
    __shared__ float outp[4][TILE][TILE];   // 4 KB    per-wave column-pass partials

    const int tid  = threadIdx.x;
    const int wave = tid >> 5;
    const int lane = tid & 31;
    const int half = (lane >> 4) & 1;       // K-half selector for WMMA fragments
    const int l    = lane & 15;             // M (A) or N (B) index within fragment

    const int img  = blockIdx.x >> 8;       // 24 images, 256 tiles each
    const int tile = blockIdx.x & 255;
    const int to0  = (tile >> 4) << 4;      // output-row tile base
    const int tw0  = (tile & 15) << 4;      // output-col tile base
    const int r0   = 4 * to0 - 5;           // input-row window base (pre-clamp)
    const int c0a  = 4 * tw0 - 8;           // ALIGNED input-col window base (= c0 - 3)

    // --- build zero-padded filter table in LDS ---
    for (int i = tid; i < KPAD_N; i += 256) kpad[i] = 0.0f;
    __syncthreads();
    if (tid < 16) {
        // K[t] = 0.25 * cubic(0.25*(7.5-t)); sums to exactly 1 (normalization identity)
        float a  = fabsf(7.5f - (float)tid) * 0.25f;
        float a2 = a * a, a3 = a2 * a;
        float c  = (a <= 1.0f) ? (1.5f * a3 - 2.5f * a2 + 1.0f)
                               : (-0.5f * a3 + 2.5f * a2 - 4.0f * a + 2.0f);
        kpad[KOFF + tid] = 0.25f * c;
    }

    // --- stage 80x84 input window into LDS ---
    const float* inimg = in + (size_t)img * (IN_H * IN_W);
    const bool interior = (r0 >= 0) && (r0 + PROWS - 1 < IN_H) &&
                          (c0a >= 0) && (c0a + PCOLS - 1 < IN_W);
    if (interior) {
        // fast path: 16B-aligned b128 loads, no clamping (~77% of tiles)
        const float* base = inimg + (size_t)r0 * IN_W + c0a;
        for (int idx = tid; idx < PROWS * NVEC; idx += 256) {
            int j = idx / NVEC;
            int q = idx - j * NVEC;
            const float* g = base + j * IN_W + 4 * q;
#if USE_ASYNC_LDS
            __builtin_amdgcn_global_load_async_to_lds_b128(
                (gv4i_p)g, (lv4i_p)&patch[j][4 * q], 0, 0);
#else
            const float4 v = *(const float4*)g;
            *(float4*)&patch[j][4 * q] = v;
#endif
        }
#if USE_ASYNC_LDS
        asm volatile("s_wait_asynccnt 0x0" ::: "memory");
#endif
    } else {
        // edge path: per-element clamped loads (keeps every LDS cell finite)
        for (int idx = tid; idx < PROWS * PCOLS; idx += 256) {
            int j = idx / PCOLS;
            int i = idx - j * PCOLS;
            int gr = clampi(r0 + j, 0, IN_H - 1);
            int gc = clampi(c0a + i, 0, IN_W - 1);
            patch[j][i] = inimg[gr * IN_W + gc];
        }
    }
    __syncthreads();

    // --- row pass: tmp[m][x] = sum_j Aband[m][j] * patch[j][x+3] ---
    // Aband[m][j] = K[j - 4m]; waves 0..4 each own a 16-wide column chunk.
    if (wave < 5) {
        const int cb = wave << 4;
        v8f acc = {};
        #pragma unroll
        for (int k = 0; k < 20; ++k) {
            const int ka = 4 * k + 2 * half;
            v2f a, b;
            a.x = kpad[ka     - 4 * l + KOFF];
            a.y = kpad[ka + 1 - 4 * l + KOFF];
            b.x = patch[ka    ][cb + l + 3];
            b.y = patch[ka + 1][cb + l + 3];
            acc = __builtin_amdgcn_wmma_f32_16x16x4_f32(
                    false, a, false, b, (short)0, acc, false, false);
        }
        #pragma unroll
        for (int r = 0; r < 8; ++r)
            tmp[r + 8 * half][cb + l] = acc[r];
    }
    __syncthreads();

    // --- column pass: out[m][n] = sum_j tmp[m][j] * Bband[j][n], Bband[j][n] = K[j - 4n] ---
    // waves 0..3 split the K dimension (5 chunks each), partials reduced below.
    if (wave < 4) {
        v8f acc = {};
        #pragma unroll
        for (int kk = 0; kk < 5; ++kk) {
            const int k  = 5 * wave + kk;
            const int ka = 4 * k + 2 * half;
            v2f a, b;
            a.x = tmp[l][ka];
            a.y = tmp[l][ka + 1];
            b.x = kpad[ka     - 4 * l + KOFF];
            b.y = kpad[ka + 1 - 4 * l + KOFF];
            acc = __builtin_amdgcn_wmma_f32_16x16x4_f32(
                    false, a, false, b, (short)0, acc, false, false);
        }
        #pragma unroll
        for (int r = 0; r < 8; ++r)
            outp[wave][r + 8 * half][l] = acc[r];
    }
    __syncthreads();

    // --- reduce 4 partial tiles, store 16x16 outputs (256 threads -> 1 pixel each) ---
    {
        const int m = tid >> 4;
        const int n = tid & 15;
        float s = outp[0][m][n] + outp[1][m][n] + outp[2][m][n] + outp[3][m][n];
        out[(size_t)img * (OUT_H * OUT_W) + (size_t)(to0 + m) * OUT_W + (tw0 + n)] = s;
    }
}

extern "C" void kernel_launch(void* const* d_in, const int* in_sizes, int n_in,
                              void* d_out, int out_size, void* d_ws, size_t ws_size,
                              hipStream_t stream) {
    (void)in_sizes; (void)n_in; (void)d_ws; (void)ws_size; (void)out_size;
    const float* in = (const float*)d_in[0];
    float* out = (float*)d_out;
    dim3 grid(NIMG * (OUT_H / TILE) * (OUT_W / TILE));   // 24 * 256 = 6144 workgroups
    bicubic_ds4_wmma<<<grid, 256, 0, stream>>>(in, out);
}